// KSpaceModule_43404939493766
// MI455X (gfx1250) — compile-verified
//
#include <hip/hip_runtime.h>
#include <cmath>

typedef __attribute__((ext_vector_type(2))) float v2f;
typedef __attribute__((ext_vector_type(8))) float v8f;
typedef __attribute__((ext_vector_type(4))) unsigned int v4u;
typedef __attribute__((ext_vector_type(8))) int v8i;
typedef __attribute__((ext_vector_type(4))) int v4i;

// workspace layout (floats)
#define WS_MESH   0     // 64: scatter accumulator
#define WS_MESHF  64    // 64: filtered mesh
#define WS_INV    128   // 9 : inv(cell), row-major

__device__ __forceinline__ float swapx16(float v) {
  // ds_swizzle 32-lane mode, xor_mask=0x10, and_mask=0x1f -> swap lane halves
  return __int_as_float(__builtin_amdgcn_ds_swizzle(__float_as_int(v), 0x401F));
}

// Branch-free tanh: prefer the gfx1250 hardware v_tanh_f32; else
// tanh(x) = 1 - 2/(exp(2x)+1) with fast exp/rcp (no EXEC divergence).
__device__ __forceinline__ float fast_tanh(float x) {
#if __has_builtin(__builtin_amdgcn_tanhf)
  return __builtin_amdgcn_tanhf(x);
#else
  float e = __expf(2.0f * x);
#if __has_builtin(__builtin_amdgcn_rcpf)
  return 1.0f - 2.0f * __builtin_amdgcn_rcpf(e + 1.0f);
#else
  return 1.0f - 2.0f / (e + 1.0f);
#endif
#endif
}

// order-3 B-spline weights + periodic indices for one dimension (NS=4)
__device__ __forceinline__ void stencil1d(float f, float w[3], int id[3]) {
  float u = f * 4.0f;
  float nr = floorf(u + 0.5f);
  float x = u - nr;
  w[0] = 0.5f * (0.5f - x) * (0.5f - x);
  w[1] = 0.75f - x * x;
  w[2] = 0.5f * (0.5f + x) * (0.5f + x);
  int n = (int)nr;
  id[0] = (n - 1) & 3; id[1] = n & 3; id[2] = (n + 1) & 3;
}

// ---------------------------------------------------------------- init
__global__ void init_kernel(const float* __restrict__ cell,
                            const float* __restrict__ b3,
                            float* __restrict__ ws, float* __restrict__ out, int N) {
  int t = threadIdx.x;
  if (t < 64) ws[WS_MESH + t] = 0.0f;
  if (t == 0) {
    float a00=cell[0],a01=cell[1],a02=cell[2];
    float a10=cell[3],a11=cell[4],a12=cell[5];
    float a20=cell[6],a21=cell[7],a22=cell[8];
    float c00 =  a11*a22 - a12*a21;
    float c01 = -(a10*a22 - a12*a20);
    float c02 =  a10*a21 - a11*a20;
    float det = a00*c00 + a01*c01 + a02*c02;
    float id  = 1.0f / det;
    ws[WS_INV+0] = c00*id;
    ws[WS_INV+1] = (a02*a21 - a01*a22)*id;
    ws[WS_INV+2] = (a01*a12 - a02*a11)*id;
    ws[WS_INV+3] = c01*id;
    ws[WS_INV+4] = (a00*a22 - a02*a20)*id;
    ws[WS_INV+5] = (a02*a10 - a00*a12)*id;
    ws[WS_INV+6] = c02*id;
    ws[WS_INV+7] = (a01*a20 - a00*a21)*id;
    ws[WS_INV+8] = (a00*a11 - a01*a10)*id;
    out[0] = (float)N * b3[0];   // fold the +b3 per particle into the init
  }
}

// ---------------------------------------------------------------- scatter
__global__ __launch_bounds__(256)
void scatter_kernel(const float* __restrict__ pos, const float* __restrict__ charges,
                    float* __restrict__ ws, int N) {
  __shared__ float lmesh[64];
  int tid = threadIdx.x;
  if (tid < 64) lmesh[tid] = 0.0f;
  float ic[9];
  #pragma unroll
  for (int i = 0; i < 9; ++i) ic[i] = ws[WS_INV + i];
  __syncthreads();

  int stride = gridDim.x * blockDim.x;
  for (int p = blockIdx.x * blockDim.x + tid; p < N; p += stride) {
    float q  = charges[p];
    float px = pos[3*p+0], py = pos[3*p+1], pz = pos[3*p+2];
    if (p + stride < N) __builtin_prefetch(&pos[3*(p+stride)], 0, 0);
    float f0 = px*ic[0] + py*ic[3] + pz*ic[6];
    float f1 = px*ic[1] + py*ic[4] + pz*ic[7];
    float f2 = px*ic[2] + py*ic[5] + pz*ic[8];
    float wx[3], wy[3], wz[3]; int ix[3], iy[3], iz[3];
    stencil1d(f0, wx, ix); stencil1d(f1, wy, iy); stencil1d(f2, wz, iz);
    #pragma unroll
    for (int a = 0; a < 3; ++a) {
      float wa = wx[a] * q; int ia = ix[a] << 4;
      #pragma unroll
      for (int b = 0; b < 3; ++b) {
        float wab = wa * wy[b]; int iab = ia | (iy[b] << 2);
        #pragma unroll
        for (int c = 0; c < 3; ++c)
          __hip_atomic_fetch_add(&lmesh[iab | iz[c]], wab * wz[c],
                                 __ATOMIC_RELAXED, __HIP_MEMORY_SCOPE_WORKGROUP);
      }
    }
  }
  __syncthreads();
  if (tid < 64)
    __hip_atomic_fetch_add(&ws[WS_MESH + tid], lmesh[tid],
                           __ATOMIC_RELAXED, __HIP_MEMORY_SCOPE_AGENT);
}

// ---------------------------------------------------------------- k-space filter
__device__ __forceinline__ float ker_half(int a, int b, int c,
                                          const float* G /*2pi*inv(cell)^T rows*/,
                                          float s2) {
  const float ftab[4] = {0.0f, 1.0f, -2.0f, -1.0f}; // fftfreq(4)*4
  float fx = ftab[a], fy = ftab[b], fz = (float)c;   // rfftfreq(4)*4 = {0,1,2}
  float k0 = fx*G[0] + fy*G[3] + fz*G[6];
  float k1 = fx*G[1] + fy*G[4] + fz*G[7];
  float k2v = fx*G[2] + fy*G[5] + fz*G[8];
  float k2 = k0*k0 + k1*k1 + k2v*k2v;
  return (k2 > 0.0f) ? __expf(-k2 * s2 * 0.5f) / k2 : 0.0f;
}

__global__ void filter_kernel(const float* __restrict__ sigma2p, float* __restrict__ ws) {
  __shared__ float sm[64], sre[64], sim[64];
  const float cs[4] = {1.0f, 0.0f, -1.0f, 0.0f};
  const float sn[4] = {0.0f, 1.0f, 0.0f, -1.0f};
  int t = threadIdx.x;             // 64 threads
  sm[t] = ws[WS_MESH + t];
  __syncthreads();
  int a = t >> 4, b = (t >> 2) & 3, c = t & 3;
  float re = 0.0f, im = 0.0f;      // forward DFT, e^{-i*2pi*k.x/4}
  for (int x = 0; x < 64; ++x) {
    int ph = (a*(x>>4) + b*((x>>2)&3) + c*(x&3)) & 3;
    float m = sm[x];
    re += m * cs[ph];
    im -= m * sn[ph];
  }
  sre[t] = re; sim[t] = im;
  float G[9];
  const float twopi = 6.2831853071795864769f;
  #pragma unroll
  for (int i = 0; i < 3; ++i)
    #pragma unroll
    for (int j = 0; j < 3; ++j)
      G[i*3+j] = twopi * ws[WS_INV + j*3 + i];
  float s2 = sigma2p[0];
  __syncthreads();
  float outv = 0.0f;               // inverse (unscaled, norm="forward")
  for (int k = 0; k < 64; ++k) {
    int ka = k >> 4, kb = (k >> 2) & 3, kc = k & 3;
    float ker = (kc <= 2) ? ker_half(ka, kb, kc, G, s2)
                          : ker_half((4-ka)&3, (4-kb)&3, 1, G, s2);
    int ph = (ka*a + kb*b + kc*c) & 3;
    outv += ker * (sre[k]*cs[ph] - sim[k]*sn[ph]);
  }
  ws[WS_MESHF + t] = outv;
}

// ---------------------------------------------------------------- gather + WMMA MLP + reduce
__global__ __launch_bounds__(256)
void gather_mlp(const float* __restrict__ pos, const float* __restrict__ charges,
                const float* __restrict__ W1, const float* __restrict__ b1,
                const float* __restrict__ W2, const float* __restrict__ b2,
                const float* __restrict__ W3,
                const float* __restrict__ ws, float* __restrict__ out, int N) {
  __shared__ float smesh[64];
  __shared__ float sred[256];
  const int tid  = threadIdx.x;
  const int lane = tid & 31;
  const int m15  = lane & 15;
  const bool lo  = lane < 16;

#if __has_builtin(__builtin_amdgcn_tensor_load_to_lds)
  // Tensor Data Mover: DMA the 64-float filtered mesh into LDS (wave 0 issues).
  if (tid < 32) {
    unsigned long long ga = (unsigned long long)(uintptr_t)(ws + WS_MESHF);
    unsigned lds = (unsigned)(uintptr_t)smesh;
    v4u g0;
    g0[0] = 1u;                                            // count=1, user mode
    g0[1] = lds;                                           // lds_addr (bytes)
    g0[2] = (unsigned)(ga & 0xffffffffu);                  // global_addr[31:0]
    g0[3] = (unsigned)((ga >> 32) & 0x01ffffffu) | (2u << 30); // addr[56:32]|type=2
    v8i g1;
    g1[0] = (int)(2u << 16);       // workgroup_mask=0, data_size=2 (4 bytes)
    g1[1] = (int)(64u << 16);      // tensor_dim0[15:0]=64 (bits 63:48 of group)
    g1[2] = (int)(1u << 16);       // tensor_dim0 hi=0 | tensor_dim1 lo=1
    g1[3] = (int)(64u << 16);      // tensor_dim1 hi=0 | tile_dim0=64
    g1[4] = 1;                     // tile_dim1=1, tile_dim2=0
    g1[5] = 64;                    // tensor_dim0_stride=64
    g1[6] = 0;
    g1[7] = 0;
    v4i gz = {0, 0, 0, 0};
#if defined(__clang_major__) && __clang_major__ >= 23
    v8i gz8 = {0, 0, 0, 0, 0, 0, 0, 0};
    __builtin_amdgcn_tensor_load_to_lds(g0, g1, gz, gz, gz8, 0);
#else
    __builtin_amdgcn_tensor_load_to_lds(g0, g1, gz, gz, 0);
#endif
    __builtin_amdgcn_s_wait_tensorcnt(0);
  }
#else
  if (tid < 64) smesh[tid] = ws[WS_MESHF + tid];
#endif
  float ic[9];
  #pragma unroll
  for (int i = 0; i < 9; ++i) ic[i] = ws[WS_INV + i];
  __syncthreads();

  // A-operands: weights transposed so M = out-feature, K = in-feature.
  v2f A1;
  A1.x = (lo && m15 < 10) ? W1[m15]      : 0.0f;   // K=0 row (charge), K=2 pad
  A1.y = (lo && m15 < 10) ? W1[10 + m15] : 0.0f;   // K=1 row (pot),    K=3 pad
  v2f A2c[3];
  {
    int koff = lo ? 0 : 2;
    #pragma unroll
    for (int cc = 0; cc < 3; ++cc) {
      int k0 = 4*cc + koff, k1 = 4*cc + 1 + koff;
      A2c[cc].x = (k0 < 10 && m15 < 10) ? W2[k0*10 + m15] : 0.0f;
      A2c[cc].y = (k1 < 10 && m15 < 10) ? W2[k1*10 + m15] : 0.0f;
    }
  }
  float b1s[8], b2s[8], w3s[8];    // D-layout lane-resident selectors
  #pragma unroll
  for (int r = 0; r < 8; ++r) {
    int f = lo ? r : r + 8;
    b1s[r] = (f < 10) ? b1[f] : 0.0f;
    b2s[r] = (f < 10) ? b2[f] : 0.0f;
    w3s[r] = (f < 10) ? W3[f] : 0.0f;
  }
  const v8f c0 = {0.0f,0.0f,0.0f,0.0f,0.0f,0.0f,0.0f,0.0f};

  const int stride   = gridDim.x * blockDim.x;
  const int wavebase = blockIdx.x * blockDim.x + (tid & ~31);
  float acc = 0.0f;

  for (int base = wavebase; base < N; base += stride) {  // wave-uniform bound
    int p  = base + lane;
    int pc = (p < N) ? p : (N - 1);
    float q  = charges[pc];
    float px = pos[3*pc+0], py = pos[3*pc+1], pz = pos[3*pc+2];
    if (base + stride < N) {                              // uniform guard
      __builtin_prefetch(&pos[3*(p + stride)], 0, 0);
      __builtin_prefetch(&charges[p + stride], 0, 0);
    }
    float f0 = px*ic[0] + py*ic[3] + pz*ic[6];
    float f1 = px*ic[1] + py*ic[4] + pz*ic[7];
    float f2 = px*ic[2] + py*ic[5] + pz*ic[8];
    float wx[3], wy[3], wz[3]; int ix[3], iy[3], iz[3];
    stencil1d(f0, wx, ix); stencil1d(f1, wy, iy); stencil1d(f2, wz, iz);
    float pot = 0.0f;
    #pragma unroll
    for (int a = 0; a < 3; ++a) {
      float wa = wx[a]; int ia = ix[a] << 4;
      #pragma unroll
      for (int b = 0; b < 3; ++b) {
        float wab = wa * wy[b]; int iab = ia | (iy[b] << 2);
        #pragma unroll
        for (int cidx = 0; cidx < 3; ++cidx)
          pot += wab * wz[cidx] * smesh[iab | iz[cidx]];
      }
    }

    float qs = swapx16(q);     // particles 16-31 into lanes 0-15
    float ps = swapx16(pot);

    #pragma unroll
    for (int g = 0; g < 2; ++g) {
      v2f B1;                  // K=4 x N=16 particles; K=2,3 zero pad
      B1.x = lo ? (g == 0 ? q   : qs) : 0.0f;
      B1.y = lo ? (g == 0 ? pot : ps) : 0.0f;
      v8f d1 = __builtin_amdgcn_wmma_f32_16x16x4_f32(
          false, A1, false, B1, (short)0, c0, false, false);
      float h[8];
      #pragma unroll
      for (int r = 0; r < 8; ++r) h[r] = fast_tanh(d1[r] + b1s[r]);

      // Re-pack h (feature-major, lane=particle) into B-layout K-chunks
      v2f B2c0, B2c1, B2c2;
      B2c0.x = lo ? h[0] : swapx16(h[2]);
      B2c0.y = lo ? h[1] : swapx16(h[3]);
      B2c1.x = lo ? h[4] : swapx16(h[6]);
      B2c1.y = lo ? h[5] : swapx16(h[7]);
      B2c2.x = lo ? swapx16(h[0]) : 0.0f;   // feature 8
      B2c2.y = lo ? swapx16(h[1]) : 0.0f;   // feature 9; K=10,11 pad
      v8f d2 = __builtin_amdgcn_wmma_f32_16x16x4_f32(
          false, A2c[2], false, B2c2, (short)0, c0, false, false);
      d2 = __builtin_amdgcn_wmma_f32_16x16x4_f32(
          false, A2c[1], false, B2c1, (short)0, d2, false, false);
      d2 = __builtin_amdgcn_wmma_f32_16x16x4_f32(
          false, A2c[0], false, B2c0, (short)0, d2, false, false);

      float accg = 0.0f;       // layer 3 + particle sum, no transpose needed
      #pragma unroll
      for (int r = 0; r < 8; ++r) accg += fast_tanh(d2[r] + b2s[r]) * w3s[r];
      float mk = ((base + g*16 + m15) < N) ? 1.0f : 0.0f;  // tail mask
      acc += accg * mk;
    }
  }

  sred[tid] = acc;
  __syncthreads();
  #pragma unroll
  for (int s = 128; s > 0; s >>= 1) {
    if (tid < s) sred[tid] += sred[tid + s];
    __syncthreads();
  }
  if (tid == 0)
    __hip_atomic_fetch_add(out, sred[0], __ATOMIC_RELAXED, __HIP_MEMORY_SCOPE_AGENT);
}

// ---------------------------------------------------------------- launch
extern "C" void kernel_launch(void* const* d_in, const int* in_sizes, int n_in,
                              void* d_out, int out_size, void* d_ws, size_t ws_size,
                              hipStream_t stream) {
  const float* positions = (const float*)d_in[0];
  const float* cell      = (const float*)d_in[1];
  const float* charges   = (const float*)d_in[2];
  const float* sigma2    = (const float*)d_in[3];
  const float* W1 = (const float*)d_in[4];
  const float* b1 = (const float*)d_in[5];
  const float* W2 = (const float*)d_in[6];
  const float* b2 = (const float*)d_in[7];
  const float* W3 = (const float*)d_in[8];
  const float* b3 = (const float*)d_in[9];
  const int N = in_sizes[2];          // charges: (N,1)
  float* ws  = (float*)d_ws;
  float* out = (float*)d_out;

  init_kernel<<<1, 64, 0, stream>>>(cell, b3, ws, out, N);
  scatter_kernel<<<2048, 256, 0, stream>>>(positions, charges, ws, N);
  filter_kernel<<<1, 64, 0, stream>>>(sigma2, ws);
  gather_mlp<<<2048, 256, 0, stream>>>(positions, charges, W1, b1, W2, b2, W3,
                                       ws, out, N);
}